// LU_1D_90142773608649
// MI455X (gfx1250) — compile-verified
//
#include <hip/hip_runtime.h>

// ---------------------------------------------------------------------------
// Hierarchical (HODLR) triangular apply:  d_out = [ L@(U@x) ; U^T@(L^T@x) ]
// All GEMM work lowered to v_wmma_f32_16x16x4_f32 (fp32 in/out, wave32).
// ---------------------------------------------------------------------------

#define MLVL   13
#define PRANK  16
#define NB     64
#define NTOT   (1 << 17)                 // 131072 rows
#define NLEAFB (1 << 13)                 // 8192 leaf blocks of 16 rows
#define CHUNK  2048                      // split-K chunk (rows)
#define TILE   (PRANK * NB)              // 1024 floats per t-tile
#define TSIZE  ((size_t)((1 << MLVL) - 1) * TILE)   // t buffer: 8,387,584 floats
#define PSLOTS 160                       // 32 partial slots per level for l=1..5
#define P1GRID 8320                      // total (level,pair,chunk) workgroups
#define RGRID  31                        // pairs in levels 1..5

typedef __attribute__((ext_vector_type(2))) float v2f;
typedef __attribute__((ext_vector_type(8))) float v8f;

struct LvlPtrs { const float* p[MLVL]; };

static __device__ inline v8f wmma4(v2f a, v2f b, v8f c) {
  // D = A(16x4,f32) * B(4x16,f32) + C(16x16,f32)
  return __builtin_amdgcn_wmma_f32_16x16x4_f32(
      /*neg_a=*/false, a, /*neg_b=*/false, b,
      /*c_mod=*/(short)0, c, /*reuse_a=*/false, /*reuse_b=*/false);
}

// ---------------------------------------------------------------------------
// Phase 1: t[l][n] (16x64) = Bm_l[n]^T (P x b) @ x_src[n] (b x 64).
// Grid flattened over (level, pair, chunk). Levels 1..5 split K across chunks
// into private partial slots (deterministic; no float atomics).
// ---------------------------------------------------------------------------
__global__ __launch_bounds__(128)
void fmm_p1(LvlPtrs Bm, const float* __restrict__ x,
            float* __restrict__ t, float* __restrict__ part, int srcHalf)
{
  int wg = blockIdx.x;
  int l = 1, b = NTOT >> 1, cppLog = 5, base = 0;
  for (int ll = 1; ll <= MLVL; ++ll) {
    int bb = NTOT >> ll;
    int cl = (ll < 6) ? (6 - ll) : 0;            // chunks-per-pair = 2^cl
    int cnt = (1 << (ll - 1)) << cl;
    if (wg < base + cnt) { l = ll; b = bb; cppLog = cl; break; }
    base += cnt;
  }
  int rel  = wg - base;
  int n    = rel >> cppLog;                      // pair index
  int c    = rel & ((1 << cppLog) - 1);          // chunk index
  int rows = (cppLog > 0) ? CHUNK : b;
  int srow0 = c * CHUNK;

  const float* Bp = Bm.p[l - 1] + ((size_t)n * b + srow0) * PRANK;          // [b,P]
  const float* Xp = x + ((size_t)(2 * n + srcHalf) * b + srow0) * NB;       // [b,64]

  const int lane = threadIdx.x & 31;
  const int lh   = lane >> 4;                    // lane-half (K group)
  const int l16  = lane & 15;
  const int cb   = (threadIdx.x >> 5) * 16;      // 4 waves -> 4 column blocks

  v8f acc = {};
  #pragma unroll 2
  for (int k = 0; k < rows; k += 4) {
    int ka = k + 2 * lh;                         // K = vgpr + 2*laneHalf
    v2f a, bv;
    a.x  = Bp[(size_t)ka * PRANK + l16];         // A[M=p][K=row] = Bm[row][p]
    a.y  = Bp[(size_t)(ka + 1) * PRANK + l16];
    bv.x = Xp[(size_t)ka * NB + cb + l16];       // B[K=row][N=col] = x[row][col]
    bv.y = Xp[(size_t)(ka + 1) * NB + cb + l16];
    acc = wmma4(a, bv, acc);
  }

  float* dst;
  if (cppLog == 0)
    dst = t + ((size_t)((1 << (l - 1)) - 1) + n) * TILE;                    // direct
  else
    dst = part + ((size_t)32 * (l - 1) + ((size_t)n << cppLog) + c) * TILE; // partial

  #pragma unroll
  for (int r = 0; r < 8; ++r)                    // C/D: row = r + 8*laneHalf
    dst[(r + 8 * lh) * NB + cb + l16] = acc[r];
}

// ---------------------------------------------------------------------------
// Phase 1b: reduce split-K partials for levels 1..5 (fixed order -> bitwise
// deterministic).
// ---------------------------------------------------------------------------
__global__ __launch_bounds__(256)
void fmm_p1reduce(const float* __restrict__ part, float* __restrict__ t)
{
  int bidx = blockIdx.x;
  int l = 1, base = 0;
  for (int ll = 1; ll <= 5; ++ll) {
    int nb2 = 1 << (ll - 1);
    if (bidx < base + nb2) { l = ll; break; }
    base += nb2;
  }
  int n = bidx - base;
  int cppLog = 6 - l, cpp = 1 << cppLog;
  const float* src = part + ((size_t)32 * (l - 1) + ((size_t)n << cppLog)) * TILE;
  float* dst = t + ((size_t)((1 << (l - 1)) - 1) + n) * TILE;
  for (int e = threadIdx.x; e < TILE; e += 256) {
    float s = 0.f;
    for (int c = 0; c < cpp; ++c) s += src[(size_t)c * TILE + e];
    dst[e] = s;
  }
}

// ---------------------------------------------------------------------------
// Phase 2: per output leaf block i (16 rows):
//   y_leaf = Dm_i @ x_leaf + sum_{levels with this leaf in tgt half}
//            A_l[rows,:] (16x16) @ t[l][n] (16x64)
// Level-membership branch is blockIdx-uniform -> EXEC stays all-ones (WMMA req).
// Safe in-place (each WG reads exactly the rows it writes).
// ---------------------------------------------------------------------------
__global__ __launch_bounds__(128)
void fmm_combine(LvlPtrs A, const float* __restrict__ D, int transD,
                 const float* __restrict__ xin, const float* __restrict__ t,
                 float* __restrict__ yout, int tgtHalf)
{
  const int i    = blockIdx.x;
  const int lane = threadIdx.x & 31;
  const int lh   = lane >> 4;
  const int l16  = lane & 15;
  const int cb   = (threadIdx.x >> 5) * 16;

  const float* Dp = D + (size_t)i * (PRANK * PRANK);
  const float* Xp = xin + (size_t)i * 16 * NB;

  v8f acc = {};
  // Leaf diagonal term: Dm (16x16) @ x_leaf (16x64)
  #pragma unroll
  for (int j = 0; j < 4; ++j) {
    int ka = 4 * j + 2 * lh;
    v2f a, bv;
    if (transD) { a.x = Dp[ka * PRANK + l16];  a.y = Dp[(ka + 1) * PRANK + l16]; }
    else        { a.x = Dp[l16 * PRANK + ka];  a.y = Dp[l16 * PRANK + ka + 1]; }
    bv.x = Xp[(size_t)ka * NB + cb + l16];
    bv.y = Xp[(size_t)(ka + 1) * NB + cb + l16];
    acc = wmma4(a, bv, acc);
  }

  // Per-level rank-P terms
  for (int l = 1; l <= MLVL; ++l) {
    int sh = MLVL - l;                       // leaf blocks per half = 2^sh
    if (((i >> sh) & 1) != tgtHalf) continue;
    int n  = i >> (sh + 1);
    int b  = NTOT >> l;
    int r0 = (i & ((1 << sh) - 1)) * 16;     // leaf row offset inside tgt half
    const float* Ap = A.p[l - 1] + ((size_t)n * b + r0) * PRANK;
    const float* Tp = t + ((size_t)((1 << (l - 1)) - 1) + n) * TILE;
    #pragma unroll
    for (int j = 0; j < 4; ++j) {
      int ka = 4 * j + 2 * lh;
      v2f a, bv;
      a.x  = Ap[l16 * PRANK + ka];           // A[M=row][K=p]
      a.y  = Ap[l16 * PRANK + ka + 1];
      bv.x = Tp[(size_t)ka * NB + cb + l16]; // B[K=p][N=col] = t[p][col]
      bv.y = Tp[(size_t)(ka + 1) * NB + cb + l16];
      acc = wmma4(a, bv, acc);
    }
  }

  float* Yp = yout + (size_t)i * 16 * NB;
  #pragma unroll
  for (int r = 0; r < 8; ++r)
    Yp[(r + 8 * lh) * NB + cb + l16] = acc[r];
}

// ---------------------------------------------------------------------------
// Host side
// ---------------------------------------------------------------------------
static void apply_factor(const LvlPtrs& A, const LvlPtrs& Bm, const float* D,
                         int transD, const float* in, float* out,
                         int srcHalf, int tgtHalf,
                         float* t, float* part, hipStream_t stream)
{
  fmm_p1<<<dim3(P1GRID), dim3(128), 0, stream>>>(Bm, in, t, part, srcHalf);
  fmm_p1reduce<<<dim3(RGRID), dim3(256), 0, stream>>>(part, t);
  fmm_combine<<<dim3(NLEAFB), dim3(128), 0, stream>>>(A, D, transD, in, t, out, tgtHalf);
}

extern "C" void kernel_launch(void* const* d_in, const int* in_sizes, int n_in,
                              void* d_out, int out_size, void* d_ws, size_t ws_size,
                              hipStream_t stream)
{
  (void)in_sizes; (void)out_size; (void)ws_size;

  const float* x = (const float*)d_in[0];
  LvlPtrs LU, LV, UU, UV;
  const float *LD, *UD;

  if (n_in >= 55) {
    // pytree-flattened: x, L_Us[0..12], L_Vs[0..12], L_D, U_Us[0..12], U_Vs[0..12], U_D
    for (int i = 0; i < MLVL; ++i) {
      LU.p[i] = (const float*)d_in[1 + i];
      LV.p[i] = (const float*)d_in[14 + i];
      UU.p[i] = (const float*)d_in[28 + i];
      UV.p[i] = (const float*)d_in[41 + i];
    }
    LD = (const float*)d_in[27];
    UD = (const float*)d_in[54];
  } else {
    // 7 inputs: per-level lists concatenated (every level has N*P/2 floats)
    const size_t lvlSz = (size_t)NTOT * PRANK / 2;   // 1,048,576 floats
    const float* lus = (const float*)d_in[1];
    const float* lvs = (const float*)d_in[2];
    const float* uus = (const float*)d_in[4];
    const float* uvs = (const float*)d_in[5];
    for (int i = 0; i < MLVL; ++i) {
      LU.p[i] = lus + (size_t)i * lvlSz;
      LV.p[i] = lvs + (size_t)i * lvlSz;
      UU.p[i] = uus + (size_t)i * lvlSz;
      UV.p[i] = uvs + (size_t)i * lvlSz;
    }
    LD = (const float*)d_in[3];
    UD = (const float*)d_in[6];
  }

  float* lo = (float*)d_out;                         // out = L@(U@x)
  float* hi = lo + (size_t)NTOT * NB;                // y = U^T@(L^T@x); also temp
  float* t    = (float*)d_ws;                        // 33.55 MB
  float* part = t + TSIZE;                           // + 0.64 MB

  // 1) tmp = U @ x          (lower=F, trans=F: A=U_Us, Bm=U_Vs, src=1, tgt=0)
  apply_factor(UU, UV, UD, 0, x,  hi, 1, 0, t, part, stream);
  // 2) out_lo = L @ tmp     (lower=T, trans=F: A=L_Us, Bm=L_Vs, src=0, tgt=1)
  apply_factor(LU, LV, LD, 0, hi, lo, 0, 1, t, part, stream);
  // 3) tmp = L^T @ x        (lower=T, trans=T: A=L_Vs, Bm=L_Us, src=1, tgt=0)
  apply_factor(LV, LU, LD, 1, x,  hi, 1, 0, t, part, stream);
  // 4) out_hi = U^T @ tmp   (lower=F, trans=T: A=U_Vs, Bm=U_Us, src=0, tgt=1) in-place
  apply_factor(UV, UU, UD, 1, hi, hi, 0, 1, t, part, stream);
}